// RecommenderModel_35493609734454
// MI455X (gfx1250) — compile-verified
//
#include <hip/hip_runtime.h>

#define U_NODES 200000
#define I_NODES 100000
#define N_NODES 300000
#define HID 64

typedef __attribute__((ext_vector_type(2))) float v2f;
typedef __attribute__((ext_vector_type(8))) float v8f;

// One wave per edge per iteration. Lane L handles columns [2L, 2L+1].
// xu/xi: source feature matrix split at the user/item boundary (for layer 0
// these are the two embedding tables; for later layers pass X and X + U*HID).
__global__ __launch_bounds__(256) void spmm_edges(
    const int* __restrict__ esrc, const int* __restrict__ edst,
    const float* __restrict__ ew,
    const float* __restrict__ xu, const float* __restrict__ xi,
    float* __restrict__ xout, int E) {
  const int lane = threadIdx.x & 31;
  const int wave = (int)((blockIdx.x * blockDim.x + threadIdx.x) >> 5);
  const int nwav = (int)((gridDim.x * blockDim.x) >> 5);
  for (int e = wave; e < E; e += nwav) {
    const int s = esrc[e];
    const int d = edst[e];
    const float w = ew[e];
    const float* row = (s < U_NODES) ? (xu + (size_t)s * HID)
                                     : (xi + (size_t)(s - U_NODES) * HID);
    // Prefetch next iteration's source row into cache (global_prefetch_b8).
    const int e2 = e + nwav;
    if (e2 < E) {
      const int s2 = esrc[e2];
      const float* row2 = (s2 < U_NODES) ? (xu + (size_t)s2 * HID)
                                         : (xi + (size_t)(s2 - U_NODES) * HID);
      __builtin_prefetch(row2 + 2 * lane, 0, 0);
    }
    const float2 v = *(const float2*)(row + 2 * lane);
    float* o = xout + (size_t)d * HID + 2 * lane;
    atomicAdd(o,     v.x * w);
    atomicAdd(o + 1, v.y * w);
  }
}

// x_final = (X0 + X1 + X2 + X3) / 4 via V_WMMA_F32_16X16X4_F32:
// K = layer axis (4), A[m,k] = 0.25 constant, B[k,n] = X_k[base+n].
// D[m,n] is the mean at position base+n, identical for all m, so the exact
// hardware (component,half)->k mapping of the 4x16 B layout is irrelevant
// (equal weights => permutation-invariant sum).
__global__ __launch_bounds__(256) void finalize_wmma(
    const float* __restrict__ ue, const float* __restrict__ ie,
    const float* __restrict__ x1, const float* __restrict__ x2,
    const float* __restrict__ x3, float* __restrict__ out) {
  const int lane = threadIdx.x & 31;
  const int wave = (int)((blockIdx.x * blockDim.x + threadIdx.x) >> 5);
  const int nwav = (int)((gridDim.x * blockDim.x) >> 5);
  const int n = lane & 15;
  const int h = lane >> 4;
  const long nchunks = (long)N_NODES * HID / 16;
  const long usplit = (long)U_NODES * HID;
  for (long c = wave; c < nchunks; c += nwav) {
    const long base = c * 16;       // 16 flat positions, all within one node row
    const long node = base / HID;   // uniform over the wave's chunk
    const long pos = base + n;
    const float* p0;
    float* po;
    if (node < U_NODES) {
      p0 = ue + pos;
      po = out + pos;                       // u_final slice
    } else {
      p0 = ie + (pos - usplit);
      po = out + 2 * usplit + (pos - usplit);  // i_final slice
    }
    v2f b;
    if (h == 0) { b.x = p0[0];   b.y = x1[pos]; }
    else        { b.x = x2[pos]; b.y = x3[pos]; }
    const v2f a = {0.25f, 0.25f};
    v8f cz = {};
    v8f d = __builtin_amdgcn_wmma_f32_16x16x4_f32(
        /*neg_a=*/false, a, /*neg_b=*/false, b,
        /*c_mod=*/(short)0, cz, /*reuse_a=*/false, /*reuse_b=*/false);
    if (lane < 16) po[0] = d[0];  // all M rows equal; lane L -> column n=L
  }
}

extern "C" void kernel_launch(void* const* d_in, const int* in_sizes, int n_in,
                              void* d_out, int out_size, void* d_ws, size_t ws_size,
                              hipStream_t stream) {
  const float* ue  = (const float*)d_in[0];
  const float* ie  = (const float*)d_in[1];
  const int* esrc  = (const int*)d_in[2];
  const int* edst  = (const int*)d_in[3];
  const float* ew  = (const float*)d_in[4];
  float* out = (float*)d_out;
  const int E = in_sizes[2];

  const size_t bufElems = (size_t)N_NODES * HID;   // 19.2M floats = 76.8 MB
  float* X1 = (float*)d_ws;
  float* X2 = X1 + bufElems;
  float* X3 = X2 + bufElems;

  // Output layout: [u_final | user_emb | i_final | item_emb]
  const size_t uElems = (size_t)U_NODES * HID;
  const size_t iElems = (size_t)I_NODES * HID;
  hipMemcpyAsync(out + uElems, ue, uElems * sizeof(float),
                 hipMemcpyDeviceToDevice, stream);
  hipMemcpyAsync(out + 2 * uElems + iElems, ie, iElems * sizeof(float),
                 hipMemcpyDeviceToDevice, stream);

  const dim3 blk(256);
  const dim3 grd(2048);

  hipMemsetAsync(X1, 0, bufElems * sizeof(float), stream);
  spmm_edges<<<grd, blk, 0, stream>>>(esrc, edst, ew, ue, ie, X1, E);

  hipMemsetAsync(X2, 0, bufElems * sizeof(float), stream);
  spmm_edges<<<grd, blk, 0, stream>>>(esrc, edst, ew, X1, X1 + uElems, X2, E);

  hipMemsetAsync(X3, 0, bufElems * sizeof(float), stream);
  spmm_edges<<<grd, blk, 0, stream>>>(esrc, edst, ew, X2, X2 + uElems, X3, E);

  finalize_wmma<<<grd, blk, 0, stream>>>(ue, ie, X1, X2, X3, out);
}